// LSTMNew_10041633538946
// MI455X (gfx1250) — compile-verified
//
#include <hip/hip_runtime.h>
#include <math.h>

// ---------------------------------------------------------------------------
// Problem constants (match reference)
// ---------------------------------------------------------------------------
#define T_STEPS 512
#define BATCH   256
#define IN_SZ   256
#define HID     100
#define GATES   (4 * HID)   // 400
#define NBLOCKS ((BATCH / 16) * ((HID + 15) / 16))   // 16 * 7 = 112

typedef float v2f __attribute__((ext_vector_type(2)));
typedef float v8f __attribute__((ext_vector_type(8)));

__device__ __forceinline__ float sigf(float x) {
    return 1.0f / (1.0f + __expf(-x));
}

// ---------------------------------------------------------------------------
// Device-wide generation barrier (all NBLOCKS single-wave blocks co-resident).
// bar[0] = arrival counter, bar[1] = generation.
// ---------------------------------------------------------------------------
__device__ __forceinline__ void grid_barrier(unsigned* bar) {
    __threadfence();   // release: make this wave's global stores visible (agent)
    if (threadIdx.x == 0) {
        unsigned* cnt = bar;
        unsigned* gen = bar + 1;
        unsigned g = __hip_atomic_load(gen, __ATOMIC_RELAXED,
                                       __HIP_MEMORY_SCOPE_AGENT);
        unsigned a = __hip_atomic_fetch_add(cnt, 1u, __ATOMIC_ACQ_REL,
                                            __HIP_MEMORY_SCOPE_AGENT);
        if (a == NBLOCKS - 1u) {
            __hip_atomic_store(cnt, 0u, __ATOMIC_RELAXED,
                               __HIP_MEMORY_SCOPE_AGENT);
            __hip_atomic_fetch_add(gen, 1u, __ATOMIC_RELEASE,
                                   __HIP_MEMORY_SCOPE_AGENT);
        } else {
            while (__hip_atomic_load(gen, __ATOMIC_ACQUIRE,
                                     __HIP_MEMORY_SCOPE_AGENT) == g) {
                __builtin_amdgcn_s_sleep(1);
            }
        }
    }
    __threadfence();   // acquire: see other blocks' stores
    __syncthreads();
}

// ---------------------------------------------------------------------------
// Kernel 1: big time-parallel GEMM  out[M,400] = X[M,K] @ W[K,400] + bias
// One wave per 16x16 output tile, V_WMMA_F32_16X16X4_F32.
// grid = (M/16, 400/16), block = 32.
// ---------------------------------------------------------------------------
__global__ __launch_bounds__(32) void gemm_xw_kernel(
    const float* __restrict__ X,     // [M, K]
    const float* __restrict__ W,     // [K, 400]
    const float* __restrict__ bias,  // [400]
    float* __restrict__ out,         // [M, 400]
    int M, int K)
{
    const int lane  = threadIdx.x;
    const int half  = lane >> 4;
    const int lm    = lane & 15;
    const int mbase = blockIdx.x * 16;
    const int n     = blockIdx.y * 16 + lm;

    v8f acc;
    const float bv = bias[n];
#pragma unroll
    for (int v = 0; v < 8; ++v) acc[v] = bv;

    const float* xrow = X + (size_t)(mbase + lm) * K;

    for (int kk = 0; kk < K; kk += 4) {
        const int k0 = kk + 2 * half;
        v2f a;                      // A (16x4): row m = mbase+lm, K pair by half
        a.x = xrow[k0];
        a.y = xrow[k0 + 1];
        v2f b;                      // B (4x16): col n, K pair by half
        b.x = W[(size_t)k0 * GATES + n];
        b.y = W[(size_t)(k0 + 1) * GATES + n];
        acc = __builtin_amdgcn_wmma_f32_16x16x4_f32(
            false, a, false, b, (short)0, acc, false, false);
    }

#pragma unroll
    for (int v = 0; v < 8; ++v) {
        const int m = mbase + v + 8 * half;
        out[(size_t)m * GATES + n] = acc[v];
    }
}

// ---------------------------------------------------------------------------
// Kernel 2: persistent LSTM layer. One launch runs all T timesteps; the 112
// single-wave blocks sync through a device-wide barrier each step.
// Block (bb, hh) owns a 16-batch x 16-hidden tile => 4 gate tiles at column
// offsets {0,100,200,300}. W_hh tile staged in LDS once, reused 512x.
// ---------------------------------------------------------------------------
__global__ __launch_bounds__(32) void lstm_layer_kernel(
    const float* __restrict__ xw,      // [T, B, 400]  x@W_ih + bias
    const float* __restrict__ w_hh,    // [100, 400]
    const float* __restrict__ h0,      // [B, 100] initial hidden (input slice)
    const float* __restrict__ c0,      // [B, 100] initial cell   (input slice)
    float* __restrict__ hb,            // [2, B, 100] ping-pong h
    float* __restrict__ cb,            // [2, B, 100] ping-pong c
    float* __restrict__ h_out,         // [T, B, 100]
    float* __restrict__ hT,            // [B, 100] final h
    float* __restrict__ cT,            // [B, 100] final c
    unsigned* __restrict__ bar)        // [2] barrier state (pre-zeroed)
{
    const int lane  = threadIdx.x;
    const int half  = lane >> 4;
    const int lm    = lane & 15;
    const int bbase = blockIdx.x * 16;        // batch tile
    const int hbase = blockIdx.y * 16;        // hidden tile
    const int hid   = hbase + lm;
    const bool hok  = (hid < HID);

    const size_t BH = (size_t)BATCH * HID;

    // Stage this block's W_hh working set into LDS: lds_w[k][gi*16+lm'].
    __shared__ float lds_w[HID * 64];         // 25.6 KB
    for (int idx = lane; idx < HID * 64; idx += 32) {
        const int k   = idx >> 6;             // 0..99
        const int j   = idx & 63;
        const int gi  = j >> 4;
        const int lmj = j & 15;
        const int hj  = hbase + lmj;
        const int col = gi * HID + (hj < HID ? hj : HID - 1);
        lds_w[idx] = w_hh[(size_t)k * GATES + col];
    }
    __syncthreads();

    for (int t = 0; t < T_STEPS; ++t) {
        const float* h_prev = (t == 0) ? h0 : hb + (size_t)(t & 1) * BH;
        const float* c_prev = (t == 0) ? c0 : cb + (size_t)(t & 1) * BH;
        float* h_next = hb + (size_t)((t + 1) & 1) * BH;
        float* c_next = cb + (size_t)((t + 1) & 1) * BH;
        const float* xwt = xw + (size_t)t * BATCH * GATES;

        // Seed accumulators with precomputed x-projection (+bias).
        v8f acc[4];
#pragma unroll
        for (int gi = 0; gi < 4; ++gi) {
#pragma unroll
            for (int v = 0; v < 8; ++v) {
                const int m = bbase + v + 8 * half;
                acc[gi][v] = hok ? xwt[(size_t)m * GATES + gi * HID + hid]
                                 : 0.0f;
            }
        }

        // Hint next step's xw slice toward the caches while WMMAs run.
        if (t + 1 < T_STEPS) {
            __builtin_prefetch(xwt + (size_t)BATCH * GATES +
                                   (size_t)(bbase + lm) * GATES + hid, 0, 1);
        }

        const float* hrow = h_prev + (size_t)(bbase + lm) * HID;

        // K loop over hidden state (H = 100, divisible by 4).
        for (int kk = 0; kk < HID; kk += 4) {
            const int k0 = kk + 2 * half;
            v2f a;
            a.x = hrow[k0];
            a.y = hrow[k0 + 1];
#pragma unroll
            for (int gi = 0; gi < 4; ++gi) {
                v2f b;
                b.x = lds_w[k0 * 64 + gi * 16 + lm];
                b.y = lds_w[(k0 + 1) * 64 + gi * 16 + lm];
                acc[gi] = __builtin_amdgcn_wmma_f32_16x16x4_f32(
                    false, a, false, b, (short)0, acc[gi], false, false);
            }
        }

        // Fused LSTM cell update.
#pragma unroll
        for (int v = 0; v < 8; ++v) {
            const int m = bbase + v + 8 * half;
            const float iv = sigf(acc[0][v]);
            const float fv = sigf(acc[1][v]);
            const float gv = tanhf(acc[2][v]);
            const float ov = sigf(acc[3][v]);
            if (hok) {
                const size_t idx = (size_t)m * HID + hid;
                const float cn = fv * c_prev[idx] + iv * gv;
                const float hn = ov * tanhf(cn);
                c_next[idx] = cn;
                h_next[idx] = hn;
                h_out[(size_t)t * BH + idx] = hn;
                if (t == T_STEPS - 1) {
                    hT[idx] = hn;
                    cT[idx] = cn;
                }
            }
        }

        // All blocks must finish step t before any block reads h/c at t+1.
        grid_barrier(bar);
    }
}

// ---------------------------------------------------------------------------
// Host orchestration
// ---------------------------------------------------------------------------
extern "C" void kernel_launch(void* const* d_in, const int* in_sizes, int n_in,
                              void* d_out, int out_size, void* d_ws, size_t ws_size,
                              hipStream_t stream) {
    (void)in_sizes; (void)n_in; (void)out_size; (void)ws_size;

    const float* x     = (const float*)d_in[0];  // [T, B, 256]
    const float* h0    = (const float*)d_in[1];  // [2, B, 100]
    const float* c0    = (const float*)d_in[2];  // [2, B, 100]
    const float* w_ih0 = (const float*)d_in[3];  // [256, 400]
    const float* w_hh0 = (const float*)d_in[4];  // [100, 400]
    const float* b0    = (const float*)d_in[5];  // [400]
    const float* w_ih1 = (const float*)d_in[6];  // [100, 400]
    const float* w_hh1 = (const float*)d_in[7];  // [100, 400]
    const float* b1    = (const float*)d_in[8];  // [400]

    float* out = (float*)d_out;
    float* ws  = (float*)d_ws;

    const size_t TB   = (size_t)T_STEPS * BATCH;   // 131072
    const size_t BH   = (size_t)BATCH * HID;       // 25600
    const size_t TB4H = TB * GATES;                // 52,428,800 floats

    // Workspace: [ xw (T*B*400) | h ping(2*BH) | c ping(2*BH) | barrier(2 u32) ]
    float*    xw  = ws;
    float*    hb  = ws + TB4H;
    float*    cb  = hb + 2 * BH;
    unsigned* bar = (unsigned*)(cb + 2 * BH);

    // Output: H [T,B,100] | h_t [2,B,100] | c_t [2,B,100]
    float* Hout = out;
    float* hT   = out + TB * HID;
    float* cT   = hT + 2 * BH;

    const dim3 blk(32, 1, 1);
    const dim3 gGemm((unsigned)(TB / 16), GATES / 16, 1);   // 8192 x 25
    const dim3 gStep(BATCH / 16, (HID + 15) / 16, 1);       // 16 x 7 = 112

    // ---------------- Layer 0 ----------------
    gemm_xw_kernel<<<gGemm, blk, 0, stream>>>(x, w_ih0, b0, xw, (int)TB, IN_SZ);
    hipMemsetAsync(bar, 0, 2 * sizeof(unsigned), stream);
    lstm_layer_kernel<<<gStep, blk, 0, stream>>>(
        xw, w_hh0, h0, c0, hb, cb,
        Hout,                 // layer-0 H staged in d_out's H region
        hT, cT, bar);

    // ---------------- Layer 1 ----------------
    // Consume layer-0 H (in d_out) before the recurrence overwrites it.
    gemm_xw_kernel<<<gGemm, blk, 0, stream>>>(Hout, w_ih1, b1, xw, (int)TB, HID);
    hipMemsetAsync(bar, 0, 2 * sizeof(unsigned), stream);
    lstm_layer_kernel<<<gStep, blk, 0, stream>>>(
        xw, w_hh1, h0 + BH, c0 + BH, hb, cb,
        Hout,                 // final H overwrites staged layer-0 H
        hT + BH, cT + BH, bar);
}